// KFFilteredPosLoss_3994319586152
// MI455X (gfx1250) — compile-verified
//
#include <hip/hip_runtime.h>

// KF filtered-position windowed MSE as a chunked linear scan:
//   kernel 1: data-independent Kalman gains kp/kv (scalar 2x2 block form)
//   kernel 2: per-chunk triangular weights W (16x16) + prefix products P
//   kernel 3: per 16-sequence tile, V_chunk = Z_chunk @ W^T + carry (x) P
//             via 4x V_WMMA_F32_16X16X4_F32 per chunk (K=16), 256 chunks
//             sequential, with global_prefetch_b8 pipelining of chunk m+1
//   kernel 4: deterministic final reduction
//
// Roofline: ~48 MB traffic (~2us @ 23.3 TB/s), ~0.1 GFLOP -> latency-bound
// sequential scan; WMMA chunking cuts the dependent chain 4096 -> 256 steps,
// prefetch overlaps HBM latency with the in-flight WMMA chain (only 64 waves,
// so there is no occupancy to hide latency otherwise).

#define DT    0.005f
#define B_SZ  512
#define T_SZ  4096
#define C_SZ  16
#define NCHUNK (T_SZ / C_SZ)   // 256
#define NSEQ  (B_SZ * 2)       // 1024 independent scalar sequences
#define NTILE (NSEQ / 16)      // 64 WMMA row-tiles

typedef float v2f __attribute__((ext_vector_type(2)));
typedef float v8f __attribute__((ext_vector_type(8)));

// -------- kernel 1: data-independent Kalman gains (2x2 per-axis block) -----
__global__ void kf_gains(const float* qp_, const float* qv_, const float* rv_,
                         float* __restrict__ kp, float* __restrict__ kv,
                         float* __restrict__ aA) {
  const float qp = qp_[0], qv = qv_[0], r = rv_[0];
  float pp = 1.f, pv = 0.f, vv = 1.f;   // P0 = I -> per-axis [[1,0],[0,1]]
  for (int t = 0; t < T_SZ; ++t) {
    const float ppp = pp + 2.f * DT * pv + DT * DT * vv + qp;
    const float pvp = pv + DT * vv;
    const float vvp = vv + qv;
    const float S   = vvp + r;
    const float kpt = pvp / S;
    const float kvt = vvp / S;
    kp[t] = kpt; kv[t] = kvt; aA[t] = 1.f - kvt;
    pp = ppp - kpt * pvp;
    pv = pvp - kpt * vvp;
    vv = vvp - kvt * vvp;
  }
}

// -------- kernel 2: per-chunk triangular weights + prefix products ---------
// v_{t0+c+1} = P[c]*v_{t0} + sum_i W[c][i]*z_{t0+i}
__global__ void kf_chunkw(const float* __restrict__ kv,
                          const float* __restrict__ aA,
                          float* __restrict__ W, float* __restrict__ P) {
  const int m = blockIdx.x * blockDim.x + threadIdx.x;
  if (m >= NCHUNK) return;
  const int t0 = m * C_SZ;
  float* Wm = W + m * C_SZ * C_SZ;
  float* Pm = P + m * C_SZ;
  float row[C_SZ];
  float prod = 1.f;
  for (int c = 0; c < C_SZ; ++c) {
    const float ac = aA[t0 + c];
    for (int i = 0; i < c; ++i) row[i] *= ac;   // row_c = a * row_{c-1}
    row[c] = kv[t0 + c];                        // + kv on diagonal
    prod *= ac;
    Pm[c] = prod;
    for (int i = 0; i < C_SZ; ++i) Wm[c * C_SZ + i] = (i <= c) ? row[i] : 0.f;
  }
}

// -------- kernel 3: WMMA chunked scan + fused loss epilogue ----------------
__global__ void __launch_bounds__(32)
kf_scan(const float* __restrict__ pred, const float* __restrict__ targ,
        const float* __restrict__ kp, const float* __restrict__ W,
        const float* __restrict__ P, float* __restrict__ partials) {
  const int tile = blockIdx.x;        // 16 sequences per tile
  const int lane = threadIdx.x;       // wave32
  const int n    = lane & 15;         // N col / A row for this lane
  const int hi   = lane >> 4;

  // A-operand addressing: row = lane&15 -> sequence; stride 2 floats in t.
  // One 128B line at (seqA, t0) covers the full 16-step chunk for that row.
  const int  seqA  = tile * 16 + n;
  const long baseA = ((long)(seqA >> 1) * T_SZ) * 2 + (seqA & 1);

  // warm the first chunk's lines before entering the dependent chain
  __builtin_prefetch(&pred[baseA], 0, 0);
  __builtin_prefetch(&targ[baseA], 0, 0);

  v8f acc = {0.f, 0.f, 0.f, 0.f, 0.f, 0.f, 0.f, 0.f};
  float lsum = 0.f;

  for (int m = 0; m < NCHUNK; ++m) {
    const int t0 = m * C_SZ;

    // pipeline: prefetch chunk m+1 operands while chunk m computes
    // (wave-uniform condition -> EXEC stays all-ones around the WMMAs)
    if (m + 1 < NCHUNK) {
      const long tnext = baseA + (long)(t0 + C_SZ) * 2;
      __builtin_prefetch(&pred[tnext], 0, 0);
      __builtin_prefetch(&targ[tnext], 0, 0);
      __builtin_prefetch(&W[(m + 1) * (C_SZ * C_SZ) + n * C_SZ], 0, 0);
    }

    // carry init: acc[j] = v_carry[M] * Pvec[n]; v_carry = prev acc col 15
    // (col 15 lives in lane 15 for rows 0-7 and lane 31 for rows 8-15)
    const float pvn = P[m * C_SZ + n];
    v8f seed;
#pragma unroll
    for (int j = 0; j < 8; ++j) {
      const float c0 = __shfl(acc[j], 15, 32);
      const float c1 = __shfl(acc[j], 31, 32);
      const float carry = hi ? c1 : c0;
      seed[j] = (m == 0) ? 0.f : carry * pvn;
    }
    acc = seed;

    // K=16 via four chained V_WMMA_F32_16X16X4_F32
    const float* Wm = W + m * (C_SZ * C_SZ);
#pragma unroll
    for (int kk = 0; kk < C_SZ; kk += 4) {
      const int k0 = kk + 2 * hi;     // A/B VGPR-pair K split across halves
      v2f a, b;
      a[0] = pred[baseA + (long)(t0 + k0) * 2];
      a[1] = pred[baseA + (long)(t0 + k0 + 1) * 2];
      b[0] = Wm[n * C_SZ + k0];       // B[k][n] = W[n][k]
      b[1] = Wm[n * C_SZ + k0 + 1];
      acc = __builtin_amdgcn_wmma_f32_16x16x4_f32(
          false, a, false, b, (short)0, acc, false, false);
    }

    // epilogue: e_s = (dt-kp_s)*v_s + kp_s*z_s - dt*w_s, s = t0 + n + 1
    const int  s     = t0 + n + 1;
    const bool valid = (s < T_SZ);
    const float kps  = kp[valid ? s : (T_SZ - 1)];
    const float wdt  = DT - kps;
#pragma unroll
    for (int j = 0; j < 8; ++j) {
      const int  M    = j + 8 * hi;
      const int  seq  = tile * 16 + M;
      const long base = ((long)(seq >> 1) * T_SZ) * 2 + (seq & 1);
      if (valid) {
        const float z  = pred[base + (long)s * 2];
        const float wv = targ[base + (long)s * 2];
        const float e  = wdt * acc[j] + kps * z - DT * wv;
        lsum += e * e;
      }
    }
  }

  // deterministic intra-wave reduction
#pragma unroll
  for (int off = 16; off > 0; off >>= 1)
    lsum += __shfl_xor(lsum, off, 32);
  if (lane == 0) partials[tile] = lsum;
}

// -------- kernel 4: deterministic final mean -------------------------------
__global__ void kf_finalize(const float* __restrict__ partials, float* out) {
  float s = 0.f;
  for (int i = 0; i < NTILE; ++i) s += partials[i];
  out[0] = s / (float)((long)B_SZ * (T_SZ - 1) * 2);
}

extern "C" void kernel_launch(void* const* d_in, const int* in_sizes, int n_in,
                              void* d_out, int out_size, void* d_ws, size_t ws_size,
                              hipStream_t stream) {
  (void)in_sizes; (void)n_in; (void)out_size; (void)ws_size;
  const float* pred = (const float*)d_in[0];  // (B,T,2)
  const float* targ = (const float*)d_in[1];  // (B,T,2)
  const float* qpos = (const float*)d_in[2];
  const float* qvel = (const float*)d_in[3];
  const float* rvel = (const float*)d_in[4];
  // d_in[5] = p0 : cancels in windowed differences, unused.

  float* ws = (float*)d_ws;
  float* kp = ws;                       // T
  float* kv = kp + T_SZ;                // T
  float* aA = kv + T_SZ;                // T
  float* W  = aA + T_SZ;                // NCHUNK*256
  float* P  = W + NCHUNK * C_SZ * C_SZ; // NCHUNK*16
  float* pr = P + NCHUNK * C_SZ;        // NTILE partials

  kf_gains<<<1, 1, 0, stream>>>(qpos, qvel, rvel, kp, kv, aA);
  kf_chunkw<<<1, NCHUNK, 0, stream>>>(kv, aA, W, P);
  kf_scan<<<NTILE, 32, 0, stream>>>(pred, targ, kp, W, P, pr);
  kf_finalize<<<1, 1, 0, stream>>>(pr, (float*)d_out);
}